// RGCN_59365037965357
// MI455X (gfx1250) — compile-verified
//
#include <hip/hip_runtime.h>

// RGCN (2-layer, mean aggr) for MI455X / gfx1250, wave32.
// - f32 WMMA 16x16x4 for all dense transforms (matches fp32 reference precision)
// - transform-then-aggregate (gather 16 floats/edge instead of 256)
// - X tiles staged to LDS via global_load_async_to_lds_b128 (ASYNCcnt)
// - counts over (relation,dst) computed once, shared by both layers
// - graph mean-pool via LDS ds_add_f32 reduction (batch is sorted)
//
// Workspace: ~30.7M floats (~123 MB).

typedef __attribute__((ext_vector_type(2))) float v2f;
typedef __attribute__((ext_vector_type(8))) float v8f;

static constexpr int N_ = 100000;   // nodes   (multiple of 16)
static constexpr int E_ = 3200000;  // edges
static constexpr int D_ = 256;      // in features
static constexpr int H_ = 16;       // hidden
static constexpr int R_ = 8;        // relations
static constexpr int C_ = 10;       // classes
static constexpr int G_ = 64;       // graphs

static constexpr int XPAD = 260;    // LDS row stride (floats): 260%64==4 -> conflict-free
                                    // b64 A-loads; 260*4 bytes %16==0 -> b128-aligned rows
static constexpr int WT1N = 9 * 16 * D_;   // transposed W1||root1: [144][256]
static constexpr int WT2N = 9 * 16 * 16;   // padded/transposed W2||root2: [9][16][16]

// ---------------------------------------------------------------- zero
__global__ void rgcn_zero4(float4* __restrict__ p, int n4) {
  int i = blockIdx.x * blockDim.x + threadIdx.x;
  if (i < n4) p[i] = make_float4(0.f, 0.f, 0.f, 0.f);
}

// ---------------------------------------------------------------- weight prep (transpose to k-contiguous)
// wt1[col][k] = (col<128 ? W1[col/16][k][col%16] : root1[k][col%16]),  col = t*16+n
// wt2[t][n][k] = (n<10 ? (t<8 ? W2[t][k][n] : root2[k][n]) : 0)
__global__ void rgcn_prepw(const float* __restrict__ W1, const float* __restrict__ root1,
                           const float* __restrict__ W2, const float* __restrict__ root2,
                           float* __restrict__ wt1, float* __restrict__ wt2) {
  int i = blockIdx.x * blockDim.x + threadIdx.x;
  if (i < WT1N) {
    int col = i >> 8, k = i & 255;
    int t = col >> 4, n = col & 15;
    const float* src = (t < 8) ? (W1 + (size_t)t * D_ * H_) : root1;
    wt1[i] = src[k * H_ + n];
  } else if (i < WT1N + WT2N) {
    int j = i - WT1N;
    int t = j >> 8, n = (j >> 4) & 15, k = j & 15;
    float v = 0.f;
    if (n < C_) v = ((t < 8) ? (W2 + (size_t)t * H_ * C_) : root2)[k * C_ + n];
    wt2[j] = v;
  }
}

// ---------------------------------------------------------------- edge counts (shared by both layers)
__global__ void rgcn_count(const int* __restrict__ ei, const int* __restrict__ ea,
                           float* __restrict__ cnt) {
  int e = blockIdx.x * blockDim.x + threadIdx.x;
  if (e >= E_) return;
  int r = ea[e];
  int d = ei[E_ + e];
  atomicAdd(cnt + (size_t)r * N_ + d, 1.0f);
}

// ---------------------------------------------------------------- layer-1 transforms
// h1rel[r][n][16] = x[n]@W1[r],  self1[n][16] = x[n]@root1 + b1
__global__ __launch_bounds__(256) void rgcn_transform1(
    const float* __restrict__ x, const float* __restrict__ wt1,
    const float* __restrict__ b1,
    float* __restrict__ h1rel, float* __restrict__ self1) {
  extern __shared__ float smem[];                  // [128][XPAD]
  const int node0 = blockIdx.x * 128;
  const unsigned ldsBase = (unsigned)(uintptr_t)(void*)smem;

  // async-stage X block: 16B per lane per op, straight to LDS (ASYNCcnt)
  for (int i = threadIdx.x; i < 128 * 64; i += 256) {
    int row = i >> 6, c4 = i & 63;
    int node = node0 + row;
    if (node < N_) {                               // OOB rows never read (16-aligned tiles)
      unsigned laddr = ldsBase + (unsigned)((row * XPAD + c4 * 4) * 4);
      const float* g = x + (size_t)node * D_ + c4 * 4;
      asm volatile("global_load_async_to_lds_b128 %0, %1, off"
                   :: "v"(laddr), "v"(g) : "memory");
    }
  }
  asm volatile("s_wait_asynccnt 0" ::: "memory");
  __syncthreads();

  const int lane = threadIdx.x & 31;
  const int half = lane >> 4;        // lanes 16-31 hold K+2,K+3
  const int mn   = lane & 15;        // m (A rows) / n (B,D cols)
  const int wv   = threadIdx.x >> 5; // 8 waves -> 8 row tiles of 16 nodes
  const int row0 = wv * 16;
  if (node0 + row0 >= N_) return;    // tile fully valid or fully OOB (N%16==0)

  const float* arow = smem + (row0 + mn) * XPAD + 2 * half;
  const int nodeb = node0 + row0 + 8 * half;       // D: VGPR v -> row nodeb+v

  for (int t = 0; t < 9; ++t) {                    // 8 relations + root
    const float* brow = wt1 + (size_t)(t * 16 + mn) * D_ + 2 * half;
    v8f acc = {};
#pragma unroll 8
    for (int k = 0; k < D_; k += 4) {
      v2f a = *(const v2f*)(arow + k);             // ds_load_b64
      v2f b = *(const v2f*)(brow + k);             // global_load_b64
      acc = __builtin_amdgcn_wmma_f32_16x16x4_f32(
          false, a, false, b, (short)0, acc, false, false);
    }
    if (t < 8) {
      float* o = h1rel + ((size_t)t * N_ + nodeb) * H_ + mn;
#pragma unroll
      for (int v = 0; v < 8; ++v) o[v * H_] = acc[v];
    } else {
      float bb = b1[mn];
      float* o = self1 + (size_t)nodeb * H_ + mn;
#pragma unroll
      for (int v = 0; v < 8; ++v) o[v * H_] = acc[v] + bb;
    }
  }
}

// ---------------------------------------------------------------- layer-1 edge scatter
__global__ void rgcn_scatter1(const int* __restrict__ ei, const int* __restrict__ ea,
                              const float* __restrict__ h1rel, float* __restrict__ sums) {
  int e = blockIdx.x * blockDim.x + threadIdx.x;
  if (e >= E_) return;
  int r = ea[e], s = ei[e], d = ei[E_ + e];
  const float4* src4 = (const float4*)(h1rel + ((size_t)r * N_ + s) * H_);
  float* drow = sums + ((size_t)r * N_ + d) * H_;
#pragma unroll
  for (int q = 0; q < 4; ++q) {
    float4 v = src4[q];
    atomicAdd(drow + q * 4 + 0, v.x);
    atomicAdd(drow + q * 4 + 1, v.y);
    atomicAdd(drow + q * 4 + 2, v.z);
    atomicAdd(drow + q * 4 + 3, v.w);
  }
}

// ---------------------------------------------------------------- layer-1 combine: mean over relations + self + relu
__global__ void rgcn_combine1(const float* __restrict__ sums, const float* __restrict__ cnt,
                              const float* __restrict__ self1, float* __restrict__ h1) {
  int i = blockIdx.x * blockDim.x + threadIdx.x;
  if (i >= N_ * H_) return;
  int n = i >> 4, f = i & 15;
  float v = self1[i];
#pragma unroll
  for (int r = 0; r < R_; ++r) {
    float c = cnt[(size_t)r * N_ + n];
    v += sums[((size_t)r * N_ + n) * H_ + f] / fmaxf(c, 1.0f);
  }
  h1[i] = fmaxf(v, 0.0f);
}

// ---------------------------------------------------------------- layer-2 transforms (K=16, B pre-padded to 16 cols)
__global__ __launch_bounds__(256) void rgcn_transform2(
    const float* __restrict__ h1, const float* __restrict__ wt2,
    const float* __restrict__ b2,
    float* __restrict__ h2rel, float* __restrict__ self2) {
  const int lane = threadIdx.x & 31;
  const int half = lane >> 4, mn = lane & 15;
  const int wv = threadIdx.x >> 5;
  const int node0 = blockIdx.x * 128 + wv * 16;
  if (node0 >= N_) return;                         // tile fully valid or fully OOB

  const float* arow = h1 + (size_t)(node0 + mn) * H_ + 2 * half;
  const float bias = (mn < C_) ? b2[mn] : 0.f;
  const int nodeb = node0 + 8 * half;

  for (int t = 0; t < 9; ++t) {
    const float* brow = wt2 + t * 256 + mn * 16 + 2 * half;
    v8f acc = {};
#pragma unroll
    for (int k = 0; k < H_; k += 4) {
      v2f a = *(const v2f*)(arow + k);
      v2f b = *(const v2f*)(brow + k);
      acc = __builtin_amdgcn_wmma_f32_16x16x4_f32(
          false, a, false, b, (short)0, acc, false, false);
    }
    if (t < 8) {
      float* o = h2rel + ((size_t)t * N_ + nodeb) * H_ + mn;
#pragma unroll
      for (int v = 0; v < 8; ++v) o[v * H_] = acc[v];
    } else {
      float* o = self2 + (size_t)nodeb * H_ + mn;
#pragma unroll
      for (int v = 0; v < 8; ++v) o[v * H_] = acc[v] + bias;
    }
  }
}

// ---------------------------------------------------------------- layer-2 edge scatter (10 features)
__global__ void rgcn_scatter2(const int* __restrict__ ei, const int* __restrict__ ea,
                              const float* __restrict__ h2rel, float* __restrict__ sums) {
  int e = blockIdx.x * blockDim.x + threadIdx.x;
  if (e >= E_) return;
  int r = ea[e], s = ei[e], d = ei[E_ + e];
  const float2* src2 = (const float2*)(h2rel + ((size_t)r * N_ + s) * H_);
  float* drow = sums + ((size_t)r * N_ + d) * H_;
#pragma unroll
  for (int q = 0; q < 5; ++q) {
    float2 v = src2[q];
    atomicAdd(drow + q * 2 + 0, v.x);
    atomicAdd(drow + q * 2 + 1, v.y);
  }
}

// ---------------------------------------------------------------- layer-2 combine -> h2[N,10]
__global__ void rgcn_combine2(const float* __restrict__ sums, const float* __restrict__ cnt,
                              const float* __restrict__ self2, float* __restrict__ h2) {
  int i = blockIdx.x * blockDim.x + threadIdx.x;
  if (i >= N_ * C_) return;
  int n = i / C_, f = i % C_;
  float v = self2[(size_t)n * H_ + f];
#pragma unroll
  for (int r = 0; r < R_; ++r) {
    float c = cnt[(size_t)r * N_ + n];
    v += sums[((size_t)r * N_ + n) * H_ + f] / fmaxf(c, 1.0f);
  }
  h2[i] = v;
}

// ---------------------------------------------------------------- graph mean pool (LDS ds_add, batch sorted)
__global__ __launch_bounds__(256) void rgcn_pool(const float* __restrict__ h2,
                                                 const int* __restrict__ batch,
                                                 float* __restrict__ psum,
                                                 float* __restrict__ pcnt) {
  __shared__ float lbuf[G_ * C_ + G_];
  for (int i = threadIdx.x; i < G_ * C_ + G_; i += 256) lbuf[i] = 0.f;
  __syncthreads();
  int n = blockIdx.x * 256 + threadIdx.x;
  if (n < N_) {
    int g = batch[n];
#pragma unroll
    for (int f = 0; f < C_; ++f)
      atomicAdd(&lbuf[g * C_ + f], h2[(size_t)n * C_ + f]);  // ds_add_f32
    atomicAdd(&lbuf[G_ * C_ + g], 1.0f);
  }
  __syncthreads();
  for (int i = threadIdx.x; i < G_ * C_; i += 256)
    if (lbuf[i] != 0.f) atomicAdd(psum + i, lbuf[i]);
  if (threadIdx.x < G_) {
    float c = lbuf[G_ * C_ + threadIdx.x];
    if (c != 0.f) atomicAdd(pcnt + threadIdx.x, c);
  }
}

__global__ void rgcn_finalize(const float* __restrict__ psum, const float* __restrict__ pcnt,
                              float* __restrict__ out) {
  int i = blockIdx.x * blockDim.x + threadIdx.x;
  if (i >= G_ * C_) return;
  out[i] = psum[i] / fmaxf(pcnt[i / C_], 1.0f);
}

// ================================================================ launch
extern "C" void kernel_launch(void* const* d_in, const int* in_sizes, int n_in,
                              void* d_out, int out_size, void* d_ws, size_t ws_size,
                              hipStream_t stream) {
  const float* x     = (const float*)d_in[0];
  const int*   ei    = (const int*)d_in[1];    // [2,E]
  const int*   ea    = (const int*)d_in[2];    // [E]
  const int*   batch = (const int*)d_in[3];    // [N]
  const float* W1    = (const float*)d_in[4];  // [R,D,H]
  const float* root1 = (const float*)d_in[5];  // [D,H]
  const float* b1    = (const float*)d_in[6];  // [H]
  const float* W2    = (const float*)d_in[7];  // [R,H,C]
  const float* root2 = (const float*)d_in[8];  // [H,C]
  const float* b2    = (const float*)d_in[9];  // [C]
  float* out = (float*)d_out;

  float* ws = (float*)d_ws;
  const size_t RNH = (size_t)R_ * N_ * H_;     // 12.8M
  float* h1rel = ws;                           // [R,N,16]
  float* self1 = h1rel + RNH;                  // [N,16]
  float* sums  = self1 + (size_t)N_ * H_;      // [R,N,16]
  float* cnt   = sums + RNH;                   // [R,N]
  float* h1    = cnt + (size_t)R_ * N_;        // [N,16]
  float* h2    = h1 + (size_t)N_ * H_;         // [N,10]
  float* psum  = h2 + (size_t)N_ * C_;         // [64*10]
  float* pcnt  = psum + G_ * C_;               // [64]
  float* wt1   = pcnt + G_;                    // [144,256]
  float* wt2   = wt1 + WT1N;                   // [9,16,16]
  float* h2rel = h1rel;                        // reuse after combine1
  float* self2 = self1;                        // reuse after combine1

  const int B = 256;
  const int nodeBlocks = (N_ + 127) / 128;     // 782
  const int edgeBlocks = (E_ + B - 1) / B;     // 12500

  rgcn_prepw<<<(WT1N + WT2N + B - 1) / B, B, 0, stream>>>(W1, root1, W2, root2, wt1, wt2);

  // layer 1
  {
    int n4 = (int)((RNH + (size_t)R_ * N_) / 4);               // sums + cnt (adjacent)
    rgcn_zero4<<<(n4 + B - 1) / B, B, 0, stream>>>((float4*)sums, n4);
  }
  rgcn_count<<<edgeBlocks, B, 0, stream>>>(ei, ea, cnt);
  rgcn_transform1<<<nodeBlocks, B, 128 * XPAD * sizeof(float), stream>>>(
      x, wt1, b1, h1rel, self1);
  rgcn_scatter1<<<edgeBlocks, B, 0, stream>>>(ei, ea, h1rel, sums);
  rgcn_combine1<<<(N_ * H_ + B - 1) / B, B, 0, stream>>>(sums, cnt, self1, h1);

  // layer 2
  {
    int n4 = (int)(RNH / 4);
    rgcn_zero4<<<(n4 + B - 1) / B, B, 0, stream>>>((float4*)sums, n4);
  }
  rgcn_transform2<<<nodeBlocks, B, 0, stream>>>(h1, wt2, b2, h2rel, self2);
  rgcn_scatter2<<<edgeBlocks, B, 0, stream>>>(ei, ea, h2rel, sums);
  rgcn_combine2<<<(N_ * C_ + B - 1) / B, B, 0, stream>>>(sums, cnt, self2, h2);

  // pooling
  rgcn_zero4<<<1, B, 0, stream>>>((float4*)psum, (G_ * C_ + G_) / 4);
  rgcn_pool<<<(N_ + 255) / 256, 256, 0, stream>>>(h2, batch, psum, pcnt);
  rgcn_finalize<<<(G_ * C_ + B - 1) / B, B, 0, stream>>>(psum, pcnt, out);
}